// Temporal_FusionBlock_83219286327327
// MI455X (gfx1250) — compile-verified
//
#include <hip/hip_runtime.h>
#include <hip/hip_bf16.h>
#include <math.h>

typedef __attribute__((ext_vector_type(2))) float v2f;
typedef __attribute__((ext_vector_type(4))) float v4f;
typedef __attribute__((ext_vector_type(8))) float v8f;

#define B_ 4
#define S_ 3
#define E_ 7
#define IMG_ 256
#define PH_ 8
#define H_ 32
#define D_ 128
#define NH_ 4
#define DK_ 64
#define DH_ 32
#define TEMB_ 256
#define LK_ 128
#define EPS_ 1e-5f
#define CIN_ 448            // E*PH*PH
#define NIMG_ 12            // B*S
#define HH_ 1024            // H*H
#define NQ_ 4096            // B*H*H

__device__ __forceinline__ v8f wmma_f32_4(v2f a, v2f b, v8f c) {
    // D = A(16x4 f32) * B(4x16 f32) + C(16x16 f32)
    return __builtin_amdgcn_wmma_f32_16x16x4_f32(false, a, false, b, (short)0, c, false, false);
}

__device__ __forceinline__ float gelu_exact(float x) {
    return 0.5f * x * (1.0f + erff(x * 0.70710678118654752f));
}

// gfx1250 async global->LDS copy (ASYNCcnt-tracked); lds_off is byte offset in
// workgroup LDS (single __shared__ array => base 0).
__device__ __forceinline__ void async_copy_b32(unsigned lds_off, const float* g) {
    asm volatile("global_load_async_to_lds_b32 %0, %1, off"
                 :: "v"(lds_off), "v"(g) : "memory");
}
__device__ __forceinline__ void wait_async0() {
    asm volatile("s_wait_asynccnt 0x0" ::: "memory");
}

// ---------------------------------------------------------------- patchify
// xp[img, c448, h1, h2] = x[b, s, e, h1*8+p1, h2*8+p2],  c448 = (p1*8+p2)*7+e
__global__ void patchify_kernel(const float* __restrict__ x, float* __restrict__ xp) {
    int idx = blockIdx.x * blockDim.x + threadIdx.x;
    if (idx >= NIMG_ * CIN_ * HH_) return;
    int p = idx & 1023;
    int rest = idx >> 10;
    int c = rest % CIN_;
    int img = rest / CIN_;
    int h1 = p >> 5, h2 = p & 31;
    int e = c % E_;
    int pp = c / E_;
    int p2 = pp & 7, p1 = pp >> 3;
    xp[idx] = x[(((size_t)img * E_ + e) * IMG_ + (h1 * PH_ + p1)) * IMG_ + (h2 * PH_ + p2)];
}

// ------------------------------------------------- patch conv (implicit GEMM, WMMA f32)
// 8 waves/block: one 16-pixel N-tile, all 128 output channels.
// B panel (448 x 16 per tap) staged in LDS via async copies, shared by all 8 waves.
#define PSTR_ 452   // padded LDS row stride (floats): 452 % 64 == 4 -> conflict-free
__global__ __launch_bounds__(256) void patch_conv_wmma(
        const float* __restrict__ xp, const float* __restrict__ pw,
        const float* __restrict__ pb, const float* __restrict__ pos,
        float* __restrict__ out) {
    __shared__ float panel[16 * PSTR_];
    int tid = threadIdx.x;
    int wave = tid >> 5, lane = tid & 31;
    int ln = lane & 15, kh = lane >> 4;
    int n0 = blockIdx.x * 16;
    int img = n0 >> 10;
    int p0 = n0 & 1023;
    // loader assignment: element (c, nL) with nL = tid&15, c = (tid>>4) + 16*j
    int nL = tid & 15;
    int cb = tid >> 4;                       // 0..15
    int pL = p0 + nL;
    int h1L = pL >> 5, h2L = pL & 31;
    // compute assignment: wave 'wave' does M-tile m0 = wave*16
    int m0 = wave * 16;
    int mA = m0 + ln;                        // A row this lane supplies
    int pD = p0 + ln;                        // D column pixel for epilogue
    v8f acc = {};
    for (int tap = 0; tap < 9; ++tap) {
        int dy = tap / 3 - 1, dx = tap % 3 - 1;
        int hh1 = h1L + dy, hh2 = h2L + dx;
        bool inb = ((unsigned)hh1 < 32u) && ((unsigned)hh2 < 32u);
        const float* gbase = xp + (size_t)img * CIN_ * HH_ + (hh1 * 32 + hh2);
        __syncthreads();                     // previous tap's compute done
        if (inb) {
            for (int j = 0; j < 28; ++j) {
                int c = cb + 16 * j;
                async_copy_b32((unsigned)(nL * PSTR_ + c) * 4u, gbase + (size_t)c * HH_);
            }
        } else {
            for (int j = 0; j < 28; ++j) {
                int c = cb + 16 * j;
                panel[nL * PSTR_ + c] = 0.0f;
            }
        }
        wait_async0();
        __syncthreads();                     // panel visible to all waves
        const float* aptr = pw + (size_t)mA * (CIN_ * 9) + tap;
        for (int c0 = 0; c0 < CIN_; c0 += 4) {
            int ca = c0 + 2 * kh;
            v2f a;
            a.x = aptr[(size_t)ca * 9];
            a.y = aptr[(size_t)(ca + 1) * 9];
            v2f bv = *(const v2f*)&panel[ln * PSTR_ + ca];   // ds_load_b64
            acc = wmma_f32_4(a, bv, acc);
        }
    }
    for (int r = 0; r < 8; ++r) {
        int mm = m0 + r + 8 * kh;
        out[((size_t)img * D_ + mm) * HH_ + pD] = acc[r] + pb[mm] + pos[(size_t)mm * HH_ + pD];
    }
}

// ---------------------------------------------------------------- group-norm stats
// group (b,g) = contiguous 24576 floats at xpc + b*393216 + g*24576
__global__ void gn_stats_kernel(const float* __restrict__ xpc, float* __restrict__ stats) {
    __shared__ float ssum[256], ssq[256];
    int bg = blockIdx.x;
    const float* base = xpc + (size_t)(bg >> 4) * 393216 + (size_t)(bg & 15) * 24576;
    float s = 0.f, q = 0.f;
    for (int i = threadIdx.x; i < 24576; i += 256) { float v = base[i]; s += v; q += v * v; }
    ssum[threadIdx.x] = s; ssq[threadIdx.x] = q;
    __syncthreads();
    for (int off = 128; off; off >>= 1) {
        if (threadIdx.x < (unsigned)off) {
            ssum[threadIdx.x] += ssum[threadIdx.x + off];
            ssq[threadIdx.x]  += ssq[threadIdx.x + off];
        }
        __syncthreads();
    }
    if (threadIdx.x == 0) {
        float mu = ssum[0] * (1.0f / 24576.0f);
        float var = ssq[0] * (1.0f / 24576.0f) - mu * mu;
        stats[bg * 2 + 0] = mu;
        stats[bg * 2 + 1] = rsqrtf(var + EPS_);
    }
}

// ---------------------------------------------------------------- t[n,d] = sum_s xn_flat[n*384+d*3+s]*w2[s]
__global__ void build_t_kernel(const float* __restrict__ xpc, const float* __restrict__ stats,
                               const float* __restrict__ gamma, const float* __restrict__ beta,
                               const float* __restrict__ w2, float* __restrict__ t) {
    int idx = blockIdx.x * blockDim.x + threadIdx.x;
    if (idx >= NQ_ * D_) return;
    int nn = idx >> 7, dd = idx & 127;
    size_t f0 = (size_t)nn * 384 + dd * 3;
    float acc = 0.f;
    for (int ss = 0; ss < 3; ++ss) {
        size_t fi = f0 + ss;
        int b = (int)(fi / 393216);
        int d2 = (int)((fi % 393216) / 3072);
        int g = d2 >> 3;
        float mu = stats[(b * 16 + g) * 2 + 0];
        float rs = stats[(b * 16 + g) * 2 + 1];
        float xv = (xpc[fi] - mu) * rs * gamma[d2] + beta[d2];
        acc += xv * w2[ss];
    }
    t[idx] = acc;
}

// ---------------------------------------------------------------- q2 = t @ W1^T (+bias), WMMA f32
__global__ void qproj_wmma(const float* __restrict__ t, const float* __restrict__ w1,
                           const float* __restrict__ b1, const float* __restrict__ w2,
                           const float* __restrict__ b2s, float* __restrict__ q2) {
    int lane = threadIdx.x;
    int ln = lane & 15, kh = lane >> 4;
    int n0 = blockIdx.x * 16, m0 = blockIdx.y * 16;
    const float* aptr = w1 + (size_t)(m0 + ln) * D_ + 2 * kh;
    const float* bptr = t  + (size_t)(n0 + ln) * D_ + 2 * kh;
    v8f acc = {};
    for (int k0 = 0; k0 < D_; k0 += 4) {
        v2f a = *(const v2f*)(aptr + k0);
        v2f b = *(const v2f*)(bptr + k0);
        acc = wmma_f32_4(a, b, acc);
    }
    float w2sum = w2[0] + w2[1] + w2[2];
    float bb = b2s[0];
    for (int r = 0; r < 8; ++r) {
        int m = m0 + r + 8 * kh;
        q2[(size_t)(n0 + ln) * 256 + m] = acc[r] + b1[m] * w2sum + bb;
    }
}

// ---------------------------------------------------------------- attention (bandwidth-bound)
// one wave per row i; k_cond/v_cond are single-use (768 MB >> 192 MB L2) -> non-temporal loads.
__global__ void attention_kernel(const float* __restrict__ qf, const float* __restrict__ kc,
                                 const float* __restrict__ vc, float* __restrict__ oc) {
    int wave = threadIdx.x >> 5, lane = threadIdx.x & 31;
    int i = blockIdx.x * 8 + wave;
    const float* q = qf + (size_t)i * DK_;
    const v4f* k4 = (const v4f*)(kc + (size_t)i * (LK_ * DK_));
    const float* v = vc + (size_t)i * (LK_ * DH_);
    float sc[4];
    for (int j = 0; j < 4; ++j) {
        int l = lane + 32 * j;
        const v4f* kr = k4 + l * (DK_ / 4);
        float s = 0.f;
        for (int d = 0; d < DK_ / 4; ++d) {
            v4f kv = __builtin_nontemporal_load(kr + d);
            v4f qv = *(const v4f*)(q + 4 * d);
            s += kv.x * qv.x + kv.y * qv.y + kv.z * qv.z + kv.w * qv.w;
        }
        sc[j] = s * 0.125f;   // 1/sqrt(64)
    }
    float m = fmaxf(fmaxf(sc[0], sc[1]), fmaxf(sc[2], sc[3]));
    for (int off = 16; off; off >>= 1) m = fmaxf(m, __shfl_xor(m, off, 32));
    float e[4];
    float sum = 0.f;
    for (int j = 0; j < 4; ++j) { e[j] = expf(sc[j] - m); sum += e[j]; }
    for (int off = 16; off; off >>= 1) sum += __shfl_xor(sum, off, 32);
    float inv = 1.0f / sum;
    for (int j = 0; j < 4; ++j) e[j] *= inv;
    float o = 0.f;
    for (int l = 0; l < LK_; ++l) {
        float a = __shfl(e[l >> 5], l & 31, 32);
        o += a * __builtin_nontemporal_load(v + (size_t)l * DH_ + lane);
    }
    oc[(size_t)(i & 4095) * 128 + (i >> 12) * 32 + lane] = o;
}

// ---------------------------------------------------------------- out-projection, WMMA f32
// o2[b,m,p] = sum_k w[m,k] * oc[b*131072 + k*1024 + p] + bias[m]
__global__ void outproj_wmma(const float* __restrict__ oc, const float* __restrict__ w,
                             const float* __restrict__ bias, float* __restrict__ o2) {
    int lane = threadIdx.x;
    int ln = lane & 15, kh = lane >> 4;
    int n = blockIdx.x * 16 + ln;
    int b = n >> 10, p = n & 1023;
    int m0 = blockIdx.y * 16;
    const float* aptr = w + (size_t)(m0 + ln) * D_ + 2 * kh;
    const float* bptr = oc + (size_t)b * 131072 + p + (size_t)(2 * kh) * HH_;
    v8f acc;
    for (int r = 0; r < 8; ++r) acc[r] = bias[m0 + r + 8 * kh];
    for (int k0 = 0; k0 < D_; k0 += 4) {
        v2f a = *(const v2f*)(aptr + k0);
        v2f bv;
        bv.x = bptr[(size_t)k0 * HH_];
        bv.y = bptr[(size_t)k0 * HH_ + HH_];
        acc = wmma_f32_4(a, bv, acc);
    }
    for (int r = 0; r < 8; ++r) {
        int m = m0 + r + 8 * kh;
        o2[((size_t)b * D_ + m) * HH_ + p] = acc[r];
    }
}

// ---------------------------------------------------------------- unpatch conv, WMMA f32
// 4 waves/block: 16-pixel N-tile, 64 of 448 output channels (grid.y = 7).
// B panel (128 x 16 per tap) staged in LDS via async copies.
#define USTR_ 132   // padded LDS row stride (floats)
__global__ __launch_bounds__(128) void unpatch_conv_wmma(
        const float* __restrict__ o2, const float* __restrict__ uw,
        const float* __restrict__ ub, float* __restrict__ u) {
    __shared__ float panel[16 * USTR_];
    int tid = threadIdx.x;
    int wave = tid >> 5, lane = tid & 31;
    int ln = lane & 15, kh = lane >> 4;
    int n0 = blockIdx.x * 16;
    int b = n0 >> 10;
    int p0 = n0 & 1023;
    int nL = tid & 15;
    int cb = tid >> 4;                       // 0..7
    int pL = p0 + nL;
    int h1L = pL >> 5, h2L = pL & 31;
    int m0 = blockIdx.y * 64 + wave * 16;
    int mA = m0 + ln;
    int pD = p0 + ln;
    v8f acc = {};
    for (int tap = 0; tap < 9; ++tap) {
        int dy = tap / 3 - 1, dx = tap % 3 - 1;
        int hh1 = h1L + dy, hh2 = h2L + dx;
        bool inb = ((unsigned)hh1 < 32u) && ((unsigned)hh2 < 32u);
        const float* gbase = o2 + (size_t)b * (D_ * HH_) + (hh1 * 32 + hh2);
        __syncthreads();
        if (inb) {
            for (int j = 0; j < 16; ++j) {
                int c = cb + 8 * j;
                async_copy_b32((unsigned)(nL * USTR_ + c) * 4u, gbase + (size_t)c * HH_);
            }
        } else {
            for (int j = 0; j < 16; ++j) {
                int c = cb + 8 * j;
                panel[nL * USTR_ + c] = 0.0f;
            }
        }
        wait_async0();
        __syncthreads();
        const float* aptr = uw + (size_t)mA * (D_ * 9) + tap;
        for (int c0 = 0; c0 < D_; c0 += 4) {
            int ca = c0 + 2 * kh;
            v2f a;
            a.x = aptr[(size_t)ca * 9];
            a.y = aptr[(size_t)(ca + 1) * 9];
            v2f bv = *(const v2f*)&panel[ln * USTR_ + ca];   // ds_load_b64
            acc = wmma_f32_4(a, bv, acc);
        }
    }
    for (int r = 0; r < 8; ++r) {
        int mm = m0 + r + 8 * kh;
        u[((size_t)b * CIN_ + mm) * HH_ + pD] = acc[r] + ub[mm];
    }
}

// ---------------------------------------------------------------- per-image stats of u
__global__ void img_stats_kernel(const float* __restrict__ u, float* __restrict__ st2) {
    __shared__ float ssum[256], ssq[256];
    int b = blockIdx.x;
    const float* base = u + (size_t)b * (CIN_ * HH_);
    float s = 0.f, q = 0.f;
    for (int i = threadIdx.x; i < CIN_ * HH_; i += 256) { float v = base[i]; s += v; q += v * v; }
    ssum[threadIdx.x] = s; ssq[threadIdx.x] = q;
    __syncthreads();
    for (int off = 128; off; off >>= 1) {
        if (threadIdx.x < (unsigned)off) {
            ssum[threadIdx.x] += ssum[threadIdx.x + off];
            ssq[threadIdx.x]  += ssq[threadIdx.x + off];
        }
        __syncthreads();
    }
    if (threadIdx.x == 0) {
        const float invN = 1.0f / (float)(CIN_ * HH_);
        float mu = ssum[0] * invN;
        float var = ssq[0] * invN - mu * mu;
        st2[b * 2 + 0] = mu;
        st2[b * 2 + 1] = rsqrtf(var + EPS_);
    }
}

// ---------------------------------------------------------------- cmod = silu(emb) @ mod_w.T + mod_b
__global__ void mod_kernel(const float* __restrict__ emb, const float* __restrict__ mw,
                           const float* __restrict__ mb, float* __restrict__ cmod) {
    int idx = threadIdx.x;
    if (idx >= B_ * 2 * E_) return;
    int b = idx / (2 * E_), o = idx % (2 * E_);
    float s = 0.f;
    for (int k = 0; k < TEMB_; ++k) {
        float e = emb[b * TEMB_ + k];
        s += (e / (1.0f + expf(-e))) * mw[o * TEMB_ + k];
    }
    cmod[idx] = s + mb[o];
}

// ---------------------------------------------------------------- final fused elementwise
__global__ void final_kernel(const float* __restrict__ u, const float* __restrict__ st2,
                             const float* __restrict__ cmod, const float* __restrict__ ew,
                             const float* __restrict__ eb, const float* __restrict__ fw,
                             const float* __restrict__ fb, float* __restrict__ out) {
    int idx = blockIdx.x * blockDim.x + threadIdx.x;
    if (idx >= B_ * IMG_ * IMG_) return;
    int b = idx >> 16;
    int pix = idx & 65535;
    int hi = pix >> 8, wi = pix & 255;
    int h1 = hi >> 3, p1 = hi & 7, h2 = wi >> 3, p2 = wi & 7;
    float mu = st2[b * 2 + 0], rs = st2[b * 2 + 1];
    float skip[E_], un[E_];
    for (int e = 0; e < E_; ++e) {
        int c = (p1 * PH_ + p2) * E_ + e;
        float v = u[((size_t)b * CIN_ + c) * HH_ + h1 * 32 + h2];
        skip[e] = v;
        float nv = (v - mu) * rs;
        un[e] = nv * (1.0f + cmod[b * 14 + e]) + cmod[b * 14 + E_ + e];
    }
    float hf[E_];
    for (int e = 0; e < E_; ++e) {
        float x1 = eb[e], x2 = eb[E_ + e];
        for (int c = 0; c < E_; ++c) {
            x1 += ew[e * E_ + c] * un[c];
            x2 += ew[(E_ + e) * E_ + c] * un[c];
        }
        hf[e] = gelu_exact(un[e]) + x1 * gelu_exact(x2);
    }
    for (int o = 0; o < E_; ++o) {
        float s = fb[o];
        for (int c = 0; c < E_; ++c) s += fw[o * E_ + c] * hf[c];
        out[((size_t)b * E_ + o) * 65536 + pix] = s + skip[o];
    }
}

extern "C" void kernel_launch(void* const* d_in, const int* in_sizes, int n_in,
                              void* d_out, int out_size, void* d_ws, size_t ws_size,
                              hipStream_t stream) {
    const float* x        = (const float*)d_in[0];
    const float* k_cond   = (const float*)d_in[1];
    const float* v_cond   = (const float*)d_in[2];
    const float* emb      = (const float*)d_in[3];
    const float* pos      = (const float*)d_in[4];
    const float* patch_w  = (const float*)d_in[5];
    const float* patch_b  = (const float*)d_in[6];
    const float* gn_gamma = (const float*)d_in[7];
    const float* gn_beta  = (const float*)d_in[8];
    const float* fc1q_w   = (const float*)d_in[9];
    const float* fc1q_b   = (const float*)d_in[10];
    const float* fc2q_w   = (const float*)d_in[11];
    const float* fc2q_b   = (const float*)d_in[12];
    const float* oproj_w  = (const float*)d_in[13];
    const float* oproj_b  = (const float*)d_in[14];
    const float* unp_w    = (const float*)d_in[15];
    const float* unp_b    = (const float*)d_in[16];
    const float* mod_w    = (const float*)d_in[17];
    const float* mod_b    = (const float*)d_in[18];
    const float* enl_w    = (const float*)d_in[19];
    const float* enl_b    = (const float*)d_in[20];
    const float* ff2_w    = (const float*)d_in[21];
    const float* ff2_b    = (const float*)d_in[22];
    float* out = (float*)d_out;

    float* ws = (float*)d_ws;
    float* xp    = ws;                         // 12*448*1024  = 5,505,024
    float* xpc   = xp    + 5505024;            // 12*128*1024  = 1,572,864
    float* tbuf  = xpc   + 1572864;            // 4096*128     =   524,288
    float* q2    = tbuf  + 524288;             // 4096*256     = 1,048,576
    float* oc    = q2    + 1048576;            // 4096*128     =   524,288
    float* o2    = oc    + 524288;             // 4*128*1024   =   524,288
    float* u     = o2    + 524288;             // 4*448*1024   = 1,835,008
    float* gnst  = u     + 1835008;            // 64*2
    float* st2   = gnst  + 128;                // 4*2
    float* cmod  = st2   + 8;                  // 4*14

    // 1. patchify
    patchify_kernel<<<(NIMG_ * CIN_ * HH_ + 255) / 256, 256, 0, stream>>>(x, xp);
    // 2. patch conv + bias + pos (WMMA, async-LDS staged): 768 N-tiles, 8 waves/block
    patch_conv_wmma<<<768, 256, 0, stream>>>(xp, patch_w, patch_b, pos, xpc);
    // 3. group-norm stats
    gn_stats_kernel<<<64, 256, 0, stream>>>(xpc, gnst);
    // 4. fused GN-normalize + fc2q contraction over s
    build_t_kernel<<<(NQ_ * D_ + 255) / 256, 256, 0, stream>>>(xpc, gnst, gn_gamma, gn_beta, fc2q_w, tbuf);
    // 5. q projection (WMMA): Ntiles=256, Mtiles=16
    qproj_wmma<<<dim3(256, 16), 32, 0, stream>>>(tbuf, fc1q_w, fc1q_b, fc2q_w, fc2q_b, q2);
    // 6. attention (bandwidth-bound NT streaming): 16384 rows / 8 waves per block
    attention_kernel<<<2048, 256, 0, stream>>>(q2, k_cond, v_cond, oc);
    // 7. output projection (WMMA): Ntiles=256, Mtiles=8
    outproj_wmma<<<dim3(256, 8), 32, 0, stream>>>(oc, oproj_w, oproj_b, o2);
    // 8. unpatch conv (WMMA, async-LDS staged): 256 N-tiles x 7 M-groups, 4 waves/block
    unpatch_conv_wmma<<<dim3(256, 7), 128, 0, stream>>>(o2, unp_w, unp_b, u);
    // 9. per-image stats
    img_stats_kernel<<<4, 256, 0, stream>>>(u, st2);
    // 10. modulation vector
    mod_kernel<<<1, 64, 0, stream>>>(emb, mod_w, mod_b, cmod);
    // 11. fused norm/mod/enlarge/gelu/ff2 + skip
    final_kernel<<<(B_ * IMG_ * IMG_ + 255) / 256, 256, 0, stream>>>(u, st2, cmod, enl_w, enl_b, ff2_w, ff2_b, out);
}